// GCN_6640019439792
// MI455X (gfx1250) — compile-verified
//
#include <hip/hip_runtime.h>
#include <hip/hip_bf16.h>
#include <math.h>

// ---------------------------------------------------------------------------
// GCN (3x GCNConv + SiLU, mean-pool, linear+relu, log_softmax) for gfx1250.
// GEMMs use V_WMMA_F32_16X16X4_F32 (exact fp32). Edge aggregation uses
// global_atomic_add_f32 against an L2-resident 25.6MB feature buffer.
// ---------------------------------------------------------------------------

typedef float v2f __attribute__((ext_vector_type(2)));
typedef float v8f __attribute__((ext_vector_type(8)));

#define HD 64   // hidden/feature dim (fixed by reference)

__device__ __forceinline__ void atomAddF(float* p, float v) {
#if defined(__AMDGCN__)
    unsafeAtomicAdd(p, v);   // lowers to global_atomic_add_f32 (no return)
#else
    atomicAdd(p, v);
#endif
}

// -------------------- degree / normalization --------------------
__global__ void k_init_deg(float* deg, int N) {
    int i = blockIdx.x * blockDim.x + threadIdx.x;
    if (i < N) deg[i] = 1.0f;               // self-loop contribution
}

__global__ void k_count_deg(const int* __restrict__ col, float* deg, int E) {
    int e = blockIdx.x * blockDim.x + threadIdx.x;
    if (e < E) atomAddF(&deg[col[e]], 1.0f);
}

__global__ void k_dinv(float* deg, int N) {
    int i = blockIdx.x * blockDim.x + threadIdx.x;
    if (i < N) deg[i] = rsqrtf(deg[i]);     // deg >= 1 always (self loops)
}

// -------------------- WMMA GEMM: XW = X[N,64] @ W[64,64] --------------------
// Block = 256 threads = 8 waves. Wave w computes rows [blockRow + 16w, +16)
// for all 64 output columns (4 N-tiles of 16), K in 16 steps of 4.
// W is staged transposed in LDS: Wl[c*64 + k] = W[k*64 + c] so each B
// fragment (two consecutive k) is a single contiguous 8-byte LDS load.
__global__ __launch_bounds__(256)
void k_gemm64(const float* __restrict__ X, const float* __restrict__ W,
              float* __restrict__ XW, int N) {
    __shared__ float Wl[HD * HD];           // 16 KB
    int t = threadIdx.x;
    for (int i = t; i < HD * HD; i += 256) {
        int c = i >> 6, k = i & 63;
        Wl[c * HD + k] = W[k * HD + c];
    }
    __syncthreads();

    int lane = t & 31;
    int wave = t >> 5;
    int m  = lane & 15;                     // matrix row (A) / col (B,C,D)
    int kh = lane >> 4;                     // K half: lanes 16-31 hold K+2,K+3

    long rowBase = (long)blockIdx.x * 128 + wave * 16;
    long rowA = rowBase + m;
    if (rowA >= N) rowA = N - 1;            // clamp: keep EXEC all-ones for WMMA
    const float* xrow = X + rowA * HD;

    v8f acc[4];
    #pragma unroll
    for (int n = 0; n < 4; ++n) acc[n] = (v8f){0,0,0,0,0,0,0,0};

    #pragma unroll
    for (int k0 = 0; k0 < HD; k0 += 4) {
        // A fragment (16x4 f32): VGPR0 = K=k0+2*kh, VGPR1 = K=k0+2*kh+1
        v2f a;
        a.x = xrow[k0 + 2 * kh + 0];
        a.y = xrow[k0 + 2 * kh + 1];
        #pragma unroll
        for (int n = 0; n < 4; ++n) {
            // B fragment (4x16 f32): lane m = col n*16+m, same K split as A
            const float* wp = &Wl[(n * 16 + m) * HD + k0 + 2 * kh];
            v2f b;
            b.x = wp[0];
            b.y = wp[1];
            acc[n] = __builtin_amdgcn_wmma_f32_16x16x4_f32(
                false, a, false, b, (short)0, acc[n], false, false);
        }
    }

    // C/D layout: VGPR r -> row (r + 8*kh), col = n*16 + m
    #pragma unroll
    for (int n = 0; n < 4; ++n) {
        #pragma unroll
        for (int r = 0; r < 8; ++r) {
            long row = rowBase + r + 8 * kh;
            if (row < N) XW[row * HD + n * 16 + m] = acc[n][r];
        }
    }
}

// -------------------- aggregation --------------------
// out[i] = xw[i] * dinv[i]^2   (self-loop message initializes the buffer)
__global__ void k_self_init(const float* __restrict__ xw,
                            const float* __restrict__ dinv,
                            float* __restrict__ out, int N) {
    long idx = (long)blockIdx.x * blockDim.x + threadIdx.x;   // over N*16
    if (idx >= (long)N * 16) return;
    long i = idx >> 4;
    int  c = (int)(idx & 15) * 4;
    float d = dinv[i];
    float s = d * d;
    float4 v = *(const float4*)(xw + i * HD + c);
    float4 o = make_float4(v.x * s, v.y * s, v.z * s, v.w * s);
    *(float4*)(out + i * HD + c) = o;
}

// out[col] += xw[row] * (dinv[row]*dinv[col]); 16 lanes/edge, float4 each
__global__ void k_scatter(const float* __restrict__ xw,
                          const int* __restrict__ row,
                          const int* __restrict__ col,
                          const float* __restrict__ dinv,
                          float* __restrict__ out, int E) {
    long idx = (long)blockIdx.x * blockDim.x + threadIdx.x;   // over E*16
    if (idx >= (long)E * 16) return;
    long e = idx >> 4;
    int  c4 = (int)(idx & 15) * 4;
    int r = row[e];
    int c = col[e];
    float nrm = dinv[r] * dinv[c];
    float4 v = *(const float4*)(xw + (long)r * HD + c4);
    float* dst = out + (long)c * HD + c4;
    atomAddF(dst + 0, v.x * nrm);
    atomAddF(dst + 1, v.y * nrm);
    atomAddF(dst + 2, v.z * nrm);
    atomAddF(dst + 3, v.w * nrm);
}

// x = silu(x + b)   (in place)
__global__ void k_bias_silu(float* __restrict__ x, const float* __restrict__ b,
                            int N) {
    long idx = (long)blockIdx.x * blockDim.x + threadIdx.x;   // over N*16
    if (idx >= (long)N * 16) return;
    long i = idx >> 4;
    int  c = (int)(idx & 15) * 4;
    float4 v = *(const float4*)(x + i * HD + c);
    float4 o;
    o.x = v.x + b[c + 0]; o.y = v.y + b[c + 1];
    o.z = v.z + b[c + 2]; o.w = v.w + b[c + 3];
    o.x = o.x / (1.0f + __expf(-o.x));
    o.y = o.y / (1.0f + __expf(-o.y));
    o.z = o.z / (1.0f + __expf(-o.z));
    o.w = o.w / (1.0f + __expf(-o.w));
    *(float4*)(x + i * HD + c) = o;
}

// -------------------- pooling + readout --------------------
__global__ void k_zero(float* p, int n) {
    int i = blockIdx.x * blockDim.x + threadIdx.x;
    if (i < n) p[i] = 0.0f;
}

__global__ void k_pool(const float* __restrict__ x, const int* __restrict__ batch,
                       float* __restrict__ pooled, float* __restrict__ cnt, int N) {
    long idx = (long)blockIdx.x * blockDim.x + threadIdx.x;   // over N*16
    if (idx >= (long)N * 16) return;
    long i = idx >> 4;
    int  c4 = (int)(idx & 15) * 4;
    int g = batch[i];
    float4 v = *(const float4*)(x + i * HD + c4);
    float* dst = pooled + (long)g * HD + c4;
    atomAddF(dst + 0, v.x);
    atomAddF(dst + 1, v.y);
    atomAddF(dst + 2, v.z);
    atomAddF(dst + 3, v.w);
    if (c4 == 0) atomAddF(&cnt[g], 1.0f);
}

// logits = relu(mean @ Wro + bro); out = log_softmax(logits)
__global__ void k_readout(const float* __restrict__ pooled,
                          const float* __restrict__ cnt,
                          const float* __restrict__ Wro,
                          const float* __restrict__ bro,
                          float* __restrict__ out, int G, int C) {
    int g = blockIdx.x * blockDim.x + threadIdx.x;
    if (g >= G) return;
    float inv = 1.0f / fmaxf(cnt[g], 1.0f);
    float logits[16];
    for (int c = 0; c < C; ++c) {
        float s = bro[c];
        for (int f = 0; f < HD; ++f)
            s += pooled[(long)g * HD + f] * inv * Wro[f * C + c];
        logits[c] = fmaxf(s, 0.0f);
    }
    float mx = -3.402823e38f;
    for (int c = 0; c < C; ++c) mx = fmaxf(mx, logits[c]);
    float se = 0.0f;
    for (int c = 0; c < C; ++c) se += __expf(logits[c] - mx);
    float lse = mx + __logf(se);
    for (int c = 0; c < C; ++c) out[(long)g * C + c] = logits[c] - lse;
}

// ---------------------------------------------------------------------------
extern "C" void kernel_launch(void* const* d_in, const int* in_sizes, int n_in,
                              void* d_out, int out_size, void* d_ws, size_t ws_size,
                              hipStream_t stream) {
    const float* x     = (const float*)d_in[0];
    const int*   ei    = (const int*)  d_in[1];
    const int*   batch = (const int*)  d_in[2];
    // d_in[3] = num_graphs (derived from out_size instead; see below)
    const float* Ws[3] = {(const float*)d_in[4], (const float*)d_in[6], (const float*)d_in[8]};
    const float* bs[3] = {(const float*)d_in[5], (const float*)d_in[7], (const float*)d_in[9]};
    const float* Wro = (const float*)d_in[10];
    const float* bro = (const float*)d_in[11];

    const int N = in_sizes[0] / HD;
    const int E = in_sizes[1] / 2;
    const int C = in_sizes[10] / HD;        // Wro is [64, C]
    const int G = out_size / C;

    const int* erow = ei;                    // edge_index[0]: sources
    const int* ecol = ei + E;                // edge_index[1]: targets

    // workspace carve-up
    float* dinv   = (float*)d_ws;
    float* bufB   = dinv + (((size_t)N + 63) & ~(size_t)63);  // xw scratch [N,64]
    float* bufA   = bufB + (size_t)N * HD;                    // agg / features [N,64]
    float* pooled = bufA + (size_t)N * HD;                    // [G,64]
    float* cnt    = pooled + (size_t)G * HD;                  // [G]

    const int T = 256;
    // --- degrees -> dinv ---
    k_init_deg <<<(N + T - 1) / T, T, 0, stream>>>(dinv, N);
    k_count_deg<<<(E + T - 1) / T, T, 0, stream>>>(ecol, dinv, E);
    k_dinv     <<<(N + T - 1) / T, T, 0, stream>>>(dinv, N);

    const long n16 = (long)N * 16;
    const long e16 = (long)E * 16;

    // --- 3 GCN layers ---
    const float* xin = x;
    for (int l = 0; l < 3; ++l) {
        k_gemm64   <<<(N + 127) / 128, 256, 0, stream>>>(xin, Ws[l], bufB, N);
        k_self_init<<<(int)((n16 + T - 1) / T), T, 0, stream>>>(bufB, dinv, bufA, N);
        k_scatter  <<<(int)((e16 + T - 1) / T), T, 0, stream>>>(bufB, erow, ecol, dinv, bufA, E);
        k_bias_silu<<<(int)((n16 + T - 1) / T), T, 0, stream>>>(bufA, bs[l], N);
        xin = bufA;
    }

    // --- mean pool + readout ---
    int pz = G * HD + G;
    k_zero   <<<(pz + T - 1) / T, T, 0, stream>>>(pooled, pz);   // pooled and cnt contiguous
    k_pool   <<<(int)((n16 + T - 1) / T), T, 0, stream>>>(bufA, batch, pooled, cnt, N);
    k_readout<<<(G + T - 1) / T, T, 0, stream>>>(pooled, cnt, Wro, bro, (float*)d_out, G, C);
}